// ChamferDistance_5669356833380
// MI455X (gfx1250) — compile-verified
//
#include <hip/hip_runtime.h>

typedef __attribute__((ext_vector_type(2))) float v2f;
typedef __attribute__((ext_vector_type(8))) float v8f;

#define NPTS   8192
#define NB     8
#define CTILES (NPTS / 16)

// D = A x B + C with A rows = (-2x,-2y,-2z,||a||^2), B cols = (x',y',z',1).
// One V_WMMA_F32_16X16X4_F32 per 16x16 tile gives ||a||^2 - 2 a.b ; add ||b||^2
// per column afterwards. Per-lane running min/argmin over column classes, final
// 16-lane shfl_xor tree per half-wave (lanes L and L+16 are independent rows).
__global__ __launch_bounds__(256) void chamfer_wmma_kernel(
    const float* __restrict__ xyz1,
    const float* __restrict__ xyz2,
    float* __restrict__ out)
{
  const int lane = threadIdx.x & 31;
  const int wave = threadIdx.x >> 5;
  const int dir  = blockIdx.z;            // 0: dist1/idx1, 1: dist2/idx2
  const int b    = blockIdx.y;
  const int rowBase = (blockIdx.x * 8 + wave) * 16;

  const float* __restrict__ Aset = (dir == 0) ? xyz1 : xyz2;
  const float* __restrict__ Bset = (dir == 0) ? xyz2 : xyz1;
  const float* Ab = Aset + (size_t)b * NPTS * 3;
  const float* Bb = Bset + (size_t)b * NPTS * 3;

  const int  lh = lane & 15;
  const bool hi = (lane >= 16);

  // A fragment (16x4): lanes 0..15 hold (K0,K1), lanes 16..31 hold (K2,K3)
  // for row rowBase+lh.  K0=-2x, K1=-2y, K2=-2z, K3=||a||^2.
  const float* ap = Ab + (size_t)(rowBase + lh) * 3;
  const float ax = ap[0], ay = ap[1], az = ap[2];
  const float asq = fmaf(ax, ax, fmaf(ay, ay, az * az));
  v2f afrag;
  afrag.x = hi ? (-2.0f * az) : (-2.0f * ax);
  afrag.y = hi ? asq          : (-2.0f * ay);

  float run_d[8];
  int   run_i[8];
#pragma unroll
  for (int s = 0; s < 8; ++s) { run_d[s] = 3.402823466e38f; run_i[s] = 0; }

#pragma unroll 2
  for (int ct = 0; ct < CTILES; ++ct) {
    const int colBase = ct * 16;
    // All 32 lanes load the full point for column colBase+lh (lanes L and
    // L+16 read the same address -> broadcast, EXEC stays full for WMMA).
    const float* bp = Bb + (size_t)(colBase + lh) * 3;
    const float bx = bp[0], by = bp[1], bz = bp[2];
    const float bsq = fmaf(bx, bx, fmaf(by, by, bz * bz));
    // B fragment (4x16) for column N = colBase+lh: K0=x', K1=y', K2=z', K3=1
    v2f bfrag;
    bfrag.x = hi ? bz   : bx;
    bfrag.y = hi ? 1.0f : by;
    v8f cz = {};
    v8f dmat = __builtin_amdgcn_wmma_f32_16x16x4_f32(
        false, afrag, false, bfrag, (short)0, cz, false, false);
    const int col = colBase + lh;
#pragma unroll
    for (int s = 0; s < 8; ++s) {
      // row M = rowBase + s + 8*hi, col N = colBase + lh
      const float dist = dmat[s] + bsq;
      if (dist < run_d[s]) { run_d[s] = dist; run_i[s] = col; }
    }
  }

  // Cross-lane min/argmin over the 16 lanes of each half-wave.
  float* distOut = out + (size_t)dir * (NB * NPTS);
  int*   idxOut  = (int*)out + 2 * (size_t)(NB * NPTS) + (size_t)dir * (NB * NPTS);
#pragma unroll
  for (int s = 0; s < 8; ++s) {
    float dmin = run_d[s];
    int   imin = run_i[s];
#pragma unroll
    for (int m = 1; m <= 8; m <<= 1) {
      const float od = __shfl_xor(dmin, m, 32);
      const int   oi = __shfl_xor(imin, m, 32);
      if (od < dmin || (od == dmin && oi < imin)) { dmin = od; imin = oi; }
    }
    if (lh == 0) {
      const int row = rowBase + s + (hi ? 8 : 0);
      distOut[(size_t)b * NPTS + row] = dmin;
      idxOut [(size_t)b * NPTS + row] = imin;
    }
  }
}

extern "C" void kernel_launch(void* const* d_in, const int* in_sizes, int n_in,
                              void* d_out, int out_size, void* d_ws, size_t ws_size,
                              hipStream_t stream) {
  const float* xyz1 = (const float*)d_in[0];
  const float* xyz2 = (const float*)d_in[1];
  float* out = (float*)d_out;
  // 512 row tiles per batch, 8 waves (1 tile each) per 256-thread block.
  dim3 grid(CTILES / 8, NB, 2);
  chamfer_wmma_kernel<<<grid, 256, 0, stream>>>(xyz1, xyz2, out);
  (void)in_sizes; (void)n_in; (void)out_size; (void)d_ws; (void)ws_size;
}